// MRConv4d_21912923144861
// MI455X (gfx1250) — compile-verified
//
#include <hip/hip_runtime.h>
#include <hip/hip_bf16.h>
#include <math.h>

typedef __attribute__((ext_vector_type(2))) float v2f;
typedef __attribute__((ext_vector_type(8))) float v8f;

extern __shared__ __align__(16) char smem_raw[];

#define NVOX 32768u        // 32*32*32 voxels per (b,c)
#define NTOT 131072u       // B * NVOX
#define CIN  256
#define COUT 128

// ---------------------------------------------------------------------------
// Kernel 1: x_j = max(0, x - min over same-parity neighbors along H,W,D)
// One block per (b,c). vol + nmin = 256KB dynamic LDS (WGP has 320KB).
// ---------------------------------------------------------------------------
__global__ void xj_kernel(const float* __restrict__ x, float* __restrict__ xj) {
    float* vol  = (float*)smem_raw;          // 32768 floats
    float* nmin = vol + 32768;               // 32768 floats
    const unsigned base = blockIdx.x * NVOX;
    const int tid = threadIdx.x;

    // cooperative load of the 32^3 volume (float4)
    for (int i = tid; i < 8192; i += 256)
        ((float4*)vol)[i] = ((const float4*)(x + base))[i];
    __syncthreads();

    const float BIG = 3.402823466e38f;

    // ---- Phase H: lines along h (stride 1024), per (w*32+d, parity) ----
    for (int t = tid; t < 2048; t += 256) {
        int p = t & 1, wd = t >> 1;
        float m1 = BIG, m2 = BIG; int a1 = -1;
        #pragma unroll
        for (int j = 0; j < 16; ++j) {
            int h = p + 2 * j;
            float v = vol[h * 1024 + wd];
            if (v < m1) { m2 = m1; m1 = v; a1 = h; } else if (v < m2) { m2 = v; }
        }
        #pragma unroll
        for (int j = 0; j < 16; ++j) {
            int h = p + 2 * j;
            nmin[h * 1024 + wd] = (h == a1) ? m2 : m1;
        }
    }
    __syncthreads();

    // ---- Phase W: lines along w (stride 32), per (h,d,parity) ----
    for (int t = tid; t < 2048; t += 256) {
        int p = t & 1, hd = t >> 1;
        int h = hd >> 5, d = hd & 31;
        int b2 = h * 1024 + d;
        float m1 = BIG, m2 = BIG; int a1 = -1;
        #pragma unroll
        for (int j = 0; j < 16; ++j) {
            int w = p + 2 * j;
            float v = vol[b2 + w * 32];
            if (v < m1) { m2 = m1; m1 = v; a1 = w; } else if (v < m2) { m2 = v; }
        }
        #pragma unroll
        for (int j = 0; j < 16; ++j) {
            int w = p + 2 * j;
            int idx = b2 + w * 32;
            nmin[idx] = fminf(nmin[idx], (w == a1) ? m2 : m1);
        }
    }
    __syncthreads();

    // ---- Phase D: lines along d (stride 1), per (h*32+w, parity) ----
    for (int t = tid; t < 2048; t += 256) {
        int p = t & 1, hw = t >> 1;
        int b2 = hw * 32;
        float m1 = BIG, m2 = BIG; int a1 = -1;
        #pragma unroll
        for (int j = 0; j < 16; ++j) {
            int d = p + 2 * j;
            float v = vol[b2 + d];
            if (v < m1) { m2 = m1; m1 = v; a1 = d; } else if (v < m2) { m2 = v; }
        }
        #pragma unroll
        for (int j = 0; j < 16; ++j) {
            int d = p + 2 * j;
            int idx = b2 + d;
            nmin[idx] = fminf(nmin[idx], (d == a1) ? m2 : m1);
        }
    }
    __syncthreads();

    // x_j = max(0, x - neighbor_min)
    for (int i = tid; i < 8192; i += 256) {
        float4 v  = ((float4*)vol)[i];
        float4 nm = ((float4*)nmin)[i];
        float4 r;
        r.x = fmaxf(0.0f, v.x - nm.x);
        r.y = fmaxf(0.0f, v.y - nm.y);
        r.z = fmaxf(0.0f, v.z - nm.z);
        r.w = fmaxf(0.0f, v.w - nm.w);
        ((float4*)(xj + base))[i] = r;
    }
}

// ---------------------------------------------------------------------------
// Kernel 2: y[o][n] = sum_c W[o][c]*xc[c][n] + bias[o] via V_WMMA_F32_16X16X4_F32
// Block = 256 threads = 8 waves; block covers 128 voxels (16 per wave),
// wave accumulates all 128 output rows (8 x v8f). conv_w staged in 128KB LDS.
// A-tile layout (ISA 7.12.2, 32-bit A 16x4): v0={K0|K2}, v1={K1|K3},
// lanes 0-15 -> M, lanes 16-31 -> same M with K+2. B mirrored over N.
// ---------------------------------------------------------------------------
__global__ void __launch_bounds__(256, 2)
gemm_kernel(const float* __restrict__ x, const float* __restrict__ xj,
            const float* __restrict__ w, const float* __restrict__ bias,
            float* __restrict__ y) {
    float* sW = (float*)smem_raw;  // 128*256 floats = 128KB
    for (int i = threadIdx.x; i < 8192; i += 256)
        ((float4*)sW)[i] = ((const float4*)w)[i];
    __syncthreads();

    const int lane = threadIdx.x & 31;
    const int wave = threadIdx.x >> 5;
    const int half = lane >> 4;      // 0: K+0/K+1 , 1: K+2/K+3
    const int l16  = lane & 15;

    const unsigned n   = blockIdx.x * 128u + wave * 16u + l16;  // voxel column
    const unsigned b   = n >> 15;
    const unsigned hwd = n & 32767u;
    const unsigned xbase = (b << 7) * NVOX + hwd;  // + c*32768

    v8f acc[8];
    #pragma unroll
    for (int m = 0; m < 8; ++m) acc[m] = (v8f)(0.0f);

    for (int k = 0; k < 64; ++k) {
        const int c0 = 4 * k + 2 * half;                  // channel pair base
        const float* src = (c0 < 128) ? x : xj;           // xc = [x ; x_j]
        const unsigned cc = (unsigned)(c0 & 127);
        v2f bv;
        bv.x = src[xbase + cc * NVOX];
        bv.y = src[xbase + cc * NVOX + NVOX];
        #pragma unroll
        for (int m = 0; m < 8; ++m) {
            const v2f av = *(const v2f*)(sW + ((m * 16 + l16) * CIN + 4 * k + 2 * half));
            acc[m] = __builtin_amdgcn_wmma_f32_16x16x4_f32(
                false, av, false, bv, (short)0, acc[m], false, false);
        }
    }

    // C/D layout: VGPR r -> M=r (lanes 0-15) / M=8+r (lanes 16-31), N=lane&15
    const unsigned ncol = blockIdx.x * 128u + wave * 16u + l16;
    #pragma unroll
    for (int m = 0; m < 8; ++m) {
        #pragma unroll
        for (int r = 0; r < 8; ++r) {
            const int o = m * 16 + half * 8 + r;
            y[(unsigned)o * NTOT + ncol] = acc[m][r] + bias[o];
        }
    }
}

// ---------------------------------------------------------------------------
// Kernel 3: per-channel sum / sumsq (deterministic tree reduce, no atomics)
// ---------------------------------------------------------------------------
__global__ void bn_stats_kernel(const float* __restrict__ y, float* __restrict__ stats) {
    const unsigned o = blockIdx.x;
    float s = 0.0f, q = 0.0f;
    for (unsigned n = threadIdx.x; n < NTOT; n += 256) {
        float v = y[o * NTOT + n];
        s += v; q += v * v;
    }
    __shared__ float ss[256], sq[256];
    ss[threadIdx.x] = s; sq[threadIdx.x] = q;
    __syncthreads();
    for (int st = 128; st > 0; st >>= 1) {
        if ((int)threadIdx.x < st) {
            ss[threadIdx.x] += ss[threadIdx.x + st];
            sq[threadIdx.x] += sq[threadIdx.x + st];
        }
        __syncthreads();
    }
    if (threadIdx.x == 0) { stats[o] = ss[0]; stats[COUT + o] = sq[0]; }
}

// ---------------------------------------------------------------------------
// Kernel 4: BN normalize + exact-erf GELU, permute [o][b*hwd] -> [b][o][hwd]
// ---------------------------------------------------------------------------
__global__ void bn_gelu_kernel(const float* __restrict__ y, const float* __restrict__ stats,
                               const float* __restrict__ gamma, const float* __restrict__ beta,
                               float* __restrict__ out) {
    const unsigned e   = (blockIdx.x * 256u + threadIdx.x) * 4u;  // flat y index
    const unsigned o   = e >> 17;
    const unsigned rem = e & (NTOT - 1u);
    const unsigned b   = rem >> 15;
    const unsigned hwd = rem & 32767u;

    const float invN = 1.0f / (float)NTOT;
    const float mean = stats[o] * invN;
    const float var  = stats[COUT + o] * invN - mean * mean;
    const float rs   = rsqrtf(var + 1e-5f);
    const float g    = gamma[o] * rs;
    const float bb   = beta[o];

    float4 v = *(const float4*)(y + e);
    float t[4] = {v.x, v.y, v.z, v.w};
    #pragma unroll
    for (int i = 0; i < 4; ++i) {
        float yn = (t[i] - mean) * g + bb;
        t[i] = 0.5f * yn * (1.0f + erff(yn * 0.70710678118654752f));
    }
    float4 r = {t[0], t[1], t[2], t[3]};
    *(float4*)(out + ((b << 7) + o) * NVOX + hwd) = r;
}

// ---------------------------------------------------------------------------
extern "C" void kernel_launch(void* const* d_in, const int* in_sizes, int n_in,
                              void* d_out, int out_size, void* d_ws, size_t ws_size,
                              hipStream_t stream) {
    const float* x      = (const float*)d_in[0];
    const float* conv_w = (const float*)d_in[1];
    const float* conv_b = (const float*)d_in[2];
    const float* gamma  = (const float*)d_in[3];
    const float* beta   = (const float*)d_in[4];
    float* out = (float*)d_out;

    char* ws = (char*)d_ws;
    float* xj    = (float*)ws;                        // 64 MiB
    float* y     = (float*)(ws + (size_t)67108864);   // 64 MiB
    float* stats = (float*)(ws + (size_t)134217728);  // 256 floats

    // 1) neighbor-diff max (one block per (b,c)); 256KB dynamic LDS
    xj_kernel<<<512, 256, 262144, stream>>>(x, xj);
    // 2) fused concat + 1x1x1 conv GEMM via f32 WMMA; 128KB LDS for conv_w
    gemm_kernel<<<1024, 256, 131072, stream>>>(x, xj, conv_w, conv_b, y);
    // 3) per-channel batch statistics
    bn_stats_kernel<<<COUT, 256, 0, stream>>>(y, stats);
    // 4) normalize + GELU + layout permute
    bn_gelu_kernel<<<16384, 256, 0, stream>>>(y, stats, gamma, beta, out);
}